// Permute_14147622273075
// MI455X (gfx1250) — compile-verified
//
#include <hip/hip_runtime.h>
#include <cstdint>

// Problem constants from the reference: NUM_P=8, B=4096, D=1024.
#define NUM_P 8
#define BATCH 4096
#define DIM   1024
#define ROWS_PER_BLOCK 16

typedef float v2f __attribute__((ext_vector_type(2)));
typedef float v4f __attribute__((ext_vector_type(4)));
typedef float v8f __attribute__((ext_vector_type(8)));

// ---------------------------------------------------------------------------
// Kernel 1: perm extraction as an exact f32 WMMA matmul:  perm = T @ iota.
// Each row of T[p] is one-hot, so sum_d T[p,o,d]*d == perm[p][o], exact in
// f32 (values < 2^10 << 2^24). One wave owns a 16-row block of one matrix and
// accumulates V_WMMA_F32_16X16X4_F32 over K=1024 (256 chunks, 2 interleaved
// accumulators to break the C-dependence chain). A-tile: one 8B NT load per
// lane per chunk (T is read exactly once, 32 MB). B-tile (iota) is built in
// registers: B[k][n] = k for all n, so every column of D carries the answer.
// ---------------------------------------------------------------------------
__global__ __launch_bounds__(256) void extract_perm_wmma(const float* __restrict__ T,
                                                         int* __restrict__ perm) {
    const int lane = threadIdx.x & 31;
    const int wave = threadIdx.x >> 5;                 // 8 waves / block
    const int gid  = blockIdx.x * 8 + wave;            // global 16-row block id
    const int p    = gid >> 6;                         // 64 row-blocks per matrix
    const int o0   = (gid & 63) * 16;

    // A layout (32-bit A 16x4): lanes 0-15 M=0..15 hold K={0,1}, lanes 16-31 K={2,3}
    const int M  = lane & 15;
    const int kb = (lane >> 4) * 2;
    const float* rowPtr = T + ((size_t)p << 20) + (size_t)(o0 + M) * DIM + kb;

    v8f c0 = {};
    v8f c1 = {};
    const float kbf = (float)kb;

    #pragma unroll 2
    for (int k0 = 0; k0 < DIM; k0 += 8) {
        // chunk A: K = k0..k0+3
        v2f a0 = __builtin_nontemporal_load((const v2f*)(rowPtr + k0));
        v2f b0;
        b0.x = (float)k0 + kbf;
        b0.y = b0.x + 1.0f;
        c0 = __builtin_amdgcn_wmma_f32_16x16x4_f32(
                 false, a0, false, b0, (short)0, c0, false, false);
        // chunk B: K = k0+4..k0+7
        v2f a1 = __builtin_nontemporal_load((const v2f*)(rowPtr + k0 + 4));
        v2f b1;
        b1.x = (float)(k0 + 4) + kbf;
        b1.y = b1.x + 1.0f;
        c1 = __builtin_amdgcn_wmma_f32_16x16x4_f32(
                 false, a1, false, b1, (short)0, c1, false, false);
    }

    // D layout: VGPR r -> (M=r, N=lane) lanes 0-15 ; (M=8+r, N=lane-16) lanes 16-31.
    // Every N column holds the same value; lanes 0 and 16 (N==0) write the 16 ints.
    if ((lane & 15) == 0) {
        const int mbase = (lane >> 4) * 8;
        int* dst = perm + (p << 10) + o0 + mbase;
        #pragma unroll
        for (int r = 0; r < 8; ++r) {
            float v = c0[r] + c1[r];              // exact: one partial is the value, other 0
            dst[r] = (int)(v + 0.5f);
        }
    }
}

// ---------------------------------------------------------------------------
// Kernel 2: out[(p*B + b), o] = x[b, perm[indices[p]][o]]
// One block = one p and a 16-row tile of b. The 4 KB perm row is staged into
// LDS via the gfx1250 async global->LDS path (ASYNCcnt), then each thread
// owns 4 consecutive output columns: 4 gathered x loads (x is L2-resident,
// 16 MB << 192 MB L2) and one coalesced non-temporal b128 store (output is
// 128 MB streaming, keep it out of L2).
// ---------------------------------------------------------------------------
__global__ __launch_bounds__(256) void gather_out(const float* __restrict__ x,
                                                  const int* __restrict__ perm,
                                                  const int* __restrict__ indices,
                                                  float* __restrict__ out) {
    __shared__ int permS[DIM];                    // 4 KB

    const int p     = blockIdx.x >> 8;            // grid = NUM_P * (BATCH/ROWS) = 8*256
    const int btile = blockIdx.x & 255;
    const int t     = threadIdx.x;                // 256 threads = 8 waves

    const int src = indices[p];                   // uniform scalar load

    // --- async copy of perm row (4 KB) into LDS: 256 lanes x 16 B ---------
    unsigned long long gbase = (unsigned long long)(const void*)(perm + (size_t)src * DIM);
    unsigned voff = (unsigned)(t * 16);                       // per-lane global byte offset
    unsigned lds  = (unsigned)(uintptr_t)permS + voff;        // LDS byte offset (low 32 bits of generic ptr)
    asm volatile("global_load_async_to_lds_b128 %0, %1, %2"
                 :: "v"(lds), "v"(voff), "s"(gbase)
                 : "memory");
    asm volatile("s_wait_asynccnt 0" ::: "memory");
    __syncthreads();

    // --- hoist this thread's 4 gather indices into registers ---------------
    const int o0 = t * 4;
    const int i0 = permS[o0 + 0];
    const int i1 = permS[o0 + 1];
    const int i2 = permS[o0 + 2];
    const int i3 = permS[o0 + 3];

    const int b0 = btile * ROWS_PER_BLOCK;
    const float* xrow = x + (size_t)b0 * DIM;
    float* orow = out + ((size_t)p * BATCH + b0) * DIM + o0;

    #pragma unroll 4
    for (int r = 0; r < ROWS_PER_BLOCK; ++r) {
        v4f v;
        v.x = xrow[i0];
        v.y = xrow[i1];
        v.z = xrow[i2];
        v.w = xrow[i3];
        __builtin_nontemporal_store(v, (v4f*)orow);   // streaming 128 MB output
        xrow += DIM;
        orow += DIM;
    }
}

// ---------------------------------------------------------------------------
// Launch wrapper. d_in order: x [B*D f32], T [NUM_P*D*D f32], indices [NUM_P i32].
// d_ws: 32 KB used for the recovered perm table.
// ---------------------------------------------------------------------------
extern "C" void kernel_launch(void* const* d_in, const int* in_sizes, int n_in,
                              void* d_out, int out_size, void* d_ws, size_t ws_size,
                              hipStream_t stream) {
    const float* x       = (const float*)d_in[0];
    const float* T       = (const float*)d_in[1];
    const int*   indices = (const int*)d_in[2];
    float*       out     = (float*)d_out;
    int*         perm    = (int*)d_ws;            // NUM_P*DIM ints = 32 KB

    // 512 waves: one per (p, 16-row block) -> 64 blocks of 8 waves.
    extract_perm_wmma<<<(NUM_P * DIM / 16) / 8, 256, 0, stream>>>(T, perm);
    gather_out<<<NUM_P * (BATCH / ROWS_PER_BLOCK), 256, 0, stream>>>(x, perm, indices, out);
}